// GNNEncoder_75342316306432
// MI455X (gfx1250) — compile-verified
//
#include <hip/hip_runtime.h>

#define GCN_D 128

typedef __attribute__((ext_vector_type(2))) float v2f;
typedef __attribute__((ext_vector_type(8))) float v8f;

// ---------------------------------------------------------------- degree prep
__global__ __launch_bounds__(256) void k_fill_deg(float* __restrict__ deg, int n) {
    int i = blockIdx.x * 256 + threadIdx.x;
    if (i < n) deg[i] = 1.0f;   // self-loop contributes 1
}

__global__ __launch_bounds__(256) void k_accum_deg(const int* __restrict__ dst,
                                                   float* __restrict__ deg, int e) {
    int i = blockIdx.x * 256 + threadIdx.x;
    if (i < e) atomicAdd(&deg[dst[i]], 1.0f);
}

__global__ __launch_bounds__(256) void k_rsqrt_inplace(float* __restrict__ deg, int n) {
    int i = blockIdx.x * 256 + threadIdx.x;
    if (i < n) deg[i] = rsqrtf(deg[i]);   // deg >= 1 always (self loops)
}

// ---------------------------------------------------------------- fp32 WMMA GEMM
// Y[N,128] = relu_opt(X[N,128]) @ W[128,128]
// Block = 256 threads = 8 waves; block owns a 16-row strip, wave w owns cols [16w,16w+16).
// A fragment (16x4 f32, ISA 7.12.2): lane L (m=L&15, kb=2*(L>>4)) holds A[m][k0+kb], A[m][k0+kb+1].
// B fragment (4x16 f32):            lane L (n=L&15, kb=2*(L>>4)) holds B[k0+kb][n], B[k0+kb+1][n].
// C/D (16x16 f32, 8 VGPRs):         vgpr r, lane L -> row r+8*(L>>4), col L&15.
__global__ __launch_bounds__(256) void k_gemm_wmma(const float* __restrict__ X,
                                                   const float* __restrict__ W,
                                                   float* __restrict__ Y,
                                                   int n, int relu_in) {
    __shared__ float Alds[16][GCN_D + 4];   // +4 floats: keeps float4 alignment, kills bank conflicts
    const int tid     = threadIdx.x;
    const int rowBase = blockIdx.x * 16;

    // Cooperative load of the 16 x 128 A strip (optionally fused ReLU).
    for (int i = tid; i < 16 * (GCN_D / 4); i += 256) {
        int r   = i >> 5;          // 32 float4 per row
        int c4  = i & 31;
        int row = rowBase + r;
        float4 v = make_float4(0.0f, 0.0f, 0.0f, 0.0f);
        if (row < n) v = ((const float4*)(X + (size_t)row * GCN_D))[c4];
        if (relu_in) {
            v.x = fmaxf(v.x, 0.0f); v.y = fmaxf(v.y, 0.0f);
            v.z = fmaxf(v.z, 0.0f); v.w = fmaxf(v.w, 0.0f);
        }
        *(float4*)&Alds[r][c4 * 4] = v;
    }
    __syncthreads();

    const int wave = tid >> 5;
    const int lane = tid & 31;
    const int n0   = wave * 16;
    const int m    = lane & 15;           // A row / B col / C col selector
    const int kb   = (lane >> 4) << 1;    // 0 or 2
    const int ncol = n0 + m;

    v8f acc = {0.f, 0.f, 0.f, 0.f, 0.f, 0.f, 0.f, 0.f};
#pragma unroll
    for (int k0 = 0; k0 < GCN_D; k0 += 4) {
        v2f a = *(const v2f*)&Alds[m][k0 + kb];                 // ds_load_b64
        v2f b;
        b.x = W[(size_t)(k0 + kb)     * GCN_D + ncol];          // coalesced 64B segments
        b.y = W[(size_t)(k0 + kb + 1) * GCN_D + ncol];
        acc = __builtin_amdgcn_wmma_f32_16x16x4_f32(
                  false, a, false, b, (short)0, acc, false, false);
    }

    const int rowOff = (lane >> 4) * 8;
#pragma unroll
    for (int r = 0; r < 8; ++r) {
        int row = rowBase + rowOff + r;
        if (row < n) Y[(size_t)row * GCN_D + ncol] = acc[r];
    }
}

// ---------------------------------------------------------------- self-loop + bias init
// out[i,:] = bias + dinv[i]^2 * xw[i,:]   (full overwrite of out)
__global__ __launch_bounds__(256) void k_self_bias(const float* __restrict__ xw,
                                                   const float* __restrict__ dinv,
                                                   const float* __restrict__ bias,
                                                   float* __restrict__ out, int n) {
    int gid = blockIdx.x * 256 + threadIdx.x;
    if (gid >= n * (GCN_D / 4)) return;
    int i    = gid >> 5;
    int lane = gid & 31;
    float di = dinv[i];
    float w  = di * di;
    float4 v  = ((const float4*)xw)[gid];
    float4 bb = ((const float4*)bias)[lane];
    float4 r;
    r.x = bb.x + w * v.x;  r.y = bb.y + w * v.y;
    r.z = bb.z + w * v.z;  r.w = bb.w + w * v.w;
    ((float4*)out)[gid] = r;
}

// ---------------------------------------------------------------- edge scatter-add
// One wave per edge; lane handles 4 consecutive floats (coalesced 512B row).
__global__ __launch_bounds__(256) void k_scatter(const float* __restrict__ xw,
                                                 const float* __restrict__ dinv,
                                                 const int* __restrict__ src,
                                                 const int* __restrict__ dst,
                                                 float* __restrict__ out, int e) {
    int gid = blockIdx.x * 256 + threadIdx.x;
    int ei  = gid >> 5;
    if (ei >= e) return;
    int lane = gid & 31;
    int s = src[ei];
    int d = dst[ei];
    float w = dinv[s] * dinv[d];
    float4 v = ((const float4*)(xw + (size_t)s * GCN_D))[lane];
    float* o = out + (size_t)d * GCN_D + lane * 4;
    atomicAdd(o + 0, w * v.x);
    atomicAdd(o + 1, w * v.y);
    atomicAdd(o + 2, w * v.z);
    atomicAdd(o + 3, w * v.w);
}

// ---------------------------------------------------------------- launcher
extern "C" void kernel_launch(void* const* d_in, const int* in_sizes, int n_in,
                              void* d_out, int out_size, void* d_ws, size_t ws_size,
                              hipStream_t stream) {
    const float* x  = (const float*)d_in[0];
    const int*  eix = (const int*)  d_in[1];
    const float* W1 = (const float*)d_in[2];
    const float* b1 = (const float*)d_in[3];
    const float* W2 = (const float*)d_in[4];
    const float* b2 = (const float*)d_in[5];
    const float* W3 = (const float*)d_in[6];
    const float* b3 = (const float*)d_in[7];
    float* out = (float*)d_out;

    const int n = in_sizes[0] / GCN_D;   // 50000
    const int e = in_sizes[1] / 2;       // 600000
    const int* srcIdx = eix;             // edge_index[0]
    const int* dstIdx = eix + e;         // edge_index[1]

    // workspace layout: dinv[N] | bufA[N*D] | bufB[N*D]   (~51.4 MB)
    float* dinv = (float*)d_ws;
    size_t pad  = ((size_t)n + 255) & ~(size_t)255;
    float* bufA = dinv + pad;
    float* bufB = bufA + (size_t)n * GCN_D;

    dim3 blk(256);
    int gN    = (n + 255) / 256;
    int gE    = (e + 255) / 256;
    int gRow  = (n + 15) / 16;
    int gElem = (n * (GCN_D / 4) + 255) / 256;
    int gScat = (e * 32 + 255) / 256;

    // degree -> dinv
    k_fill_deg     <<<gN, blk, 0, stream>>>(dinv, n);
    k_accum_deg    <<<gE, blk, 0, stream>>>(dstIdx, dinv, e);
    k_rsqrt_inplace<<<gN, blk, 0, stream>>>(dinv, n);

    // layer 1
    k_gemm_wmma<<<gRow,  blk, 0, stream>>>(x,    W1, bufA, n, 0);
    k_self_bias<<<gElem, blk, 0, stream>>>(bufA, dinv, b1, bufB, n);
    k_scatter  <<<gScat, blk, 0, stream>>>(bufA, dinv, srcIdx, dstIdx, bufB, e);
    // layer 2 (ReLU fused into GEMM input read)
    k_gemm_wmma<<<gRow,  blk, 0, stream>>>(bufB, W2, bufA, n, 1);
    k_self_bias<<<gElem, blk, 0, stream>>>(bufA, dinv, b2, bufB, n);
    k_scatter  <<<gScat, blk, 0, stream>>>(bufA, dinv, srcIdx, dstIdx, bufB, e);
    // layer 3 (no output ReLU) -> d_out
    k_gemm_wmma<<<gRow,  blk, 0, stream>>>(bufB, W3, bufA, n, 1);
    k_self_bias<<<gElem, blk, 0, stream>>>(bufA, dinv, b3, out, n);
    k_scatter  <<<gScat, blk, 0, stream>>>(bufA, dinv, srcIdx, dstIdx, out, e);
}